// CostLearning_3315714753077
// MI455X (gfx1250) — compile-verified
//
#include <hip/hip_runtime.h>

typedef __attribute__((ext_vector_type(2))) float v2f;
typedef __attribute__((ext_vector_type(8))) float v8f;

#define BT_TOTAL (128 * 2048)
#define D_S 128
#define D_A 32

// One wave handles 16 rows. A(16x4) = squared inputs, B(4x16) = exp-weights
// replicated across all 16 columns, C/D accumulates f32 row sums across the
// k-chunks. All columns of D are identical, so lane 0 / lane 16 hold the
// 16 per-row results in their 8 accumulator VGPRs (ISA 7.12.2 C/D layout).
__global__ __launch_bounds__(256) void cost_wmma_kernel(
    const float* __restrict__ states,   // [BT, 128]
    const float* __restrict__ actions,  // [BT, 32]
    const float* __restrict__ q_diag_log,  // [128]
    const float* __restrict__ r_diag_log,  // [32]
    float* __restrict__ out)            // [BT]
{
    __shared__ float w[D_S + D_A];

    const int tid = threadIdx.x;
    if (tid < D_S) {
        w[tid] = __expf(q_diag_log[tid]);
    } else if (tid < D_S + D_A) {
        w[tid] = __expf(r_diag_log[tid - D_S]);
    }
    __syncthreads();

    const int lane   = tid & 31;
    const int waveId = tid >> 5;
    const int half   = lane >> 4;   // 0: K pair (0,1)   1: K pair (2,3)
    const int m      = lane & 15;   // row within 16-row tile

    const int row0 = (blockIdx.x * 8 + waveId) * 16;
    const int r    = row0 + m;

    const float* sp = states  + (size_t)r * D_S + half * 2;
    const float* ap = actions + (size_t)r * D_A + half * 2;
    const float* wq = w        + half * 2;
    const float* wr = w + D_S  + half * 2;

    v8f acc = {0.f, 0.f, 0.f, 0.f, 0.f, 0.f, 0.f, 0.f};

    // States: 128 = 32 chunks of K=4
#pragma unroll
    for (int c = 0; c < D_S / 4; ++c) {
        v2f s = *(const v2f*)(sp + 4 * c);   // global_load_b64, coalesced across tile
        v2f b = *(const v2f*)(wq + 4 * c);   // ds_load_b64 (exp weights)
        v2f a;
        a.x = s.x * s.x;
        a.y = s.y * s.y;
        acc = __builtin_amdgcn_wmma_f32_16x16x4_f32(
            /*neg_a=*/false, a, /*neg_b=*/false, b,
            /*c_mod=*/(short)0, acc, /*reuse_a=*/false, /*reuse_b=*/false);
    }

    // Actions: 32 = 8 chunks of K=4
#pragma unroll
    for (int c = 0; c < D_A / 4; ++c) {
        v2f s = *(const v2f*)(ap + 4 * c);
        v2f b = *(const v2f*)(wr + 4 * c);
        v2f a;
        a.x = s.x * s.x;
        a.y = s.y * s.y;
        acc = __builtin_amdgcn_wmma_f32_16x16x4_f32(
            false, a, false, b, (short)0, acc, false, false);
    }

    // D layout: VGPR j = row (j + 8*half) of the tile, column = lane%16.
    // Every column is identical, so lanes with (lane%16)==0 own valid data:
    // lane 0 -> rows 0..7, lane 16 -> rows 8..15.
    if (m == 0) {
        *(v8f*)(out + row0 + half * 8) = acc;   // 32B-aligned, 2x global_store_b128
    }
}

extern "C" void kernel_launch(void* const* d_in, const int* in_sizes, int n_in,
                              void* d_out, int out_size, void* d_ws, size_t ws_size,
                              hipStream_t stream) {
    const float* states     = (const float*)d_in[0];
    const float* actions    = (const float*)d_in[1];
    const float* q_diag_log = (const float*)d_in[2];
    const float* r_diag_log = (const float*)d_in[3];
    float* out = (float*)d_out;

    // 256 threads = 8 waves, 16 rows/wave -> 128 rows per block
    const int rowsPerBlock = 128;
    const int nBlocks = BT_TOTAL / rowsPerBlock;  // 2048
    cost_wmma_kernel<<<nBlocks, 256, 0, stream>>>(states, actions, q_diag_log,
                                                  r_diag_log, out);
}